// FeatureTransformer_17454747091331
// MI455X (gfx1250) — compile-verified
//
#include <hip/hip_runtime.h>
#include <hip/hip_bf16.h>

// Problem constants (match reference).
#define N_POS   4096
#define KACT    32
#define DFEAT   49152
#define BASEDIM 512
#define F0DIM   768
#define F1DIM   64

#define CSPLIT   2                   // column split across blockIdx.y
#define HALFCOLS (BASEDIM / CSPLIT)  // 256 columns per block
#define BSTRIDE  272                 // padded LDS row stride (272 % 64 == 16 -> half-waves hit disjoint banks)
#define NBUF     3                   // triple buffer: depth-2 async prefetch

typedef __attribute__((ext_vector_type(2))) float v2f;
typedef __attribute__((ext_vector_type(8))) float v8f;

// CDNA5 async copy: 16 bytes per lane, global memory -> LDS, no VGPR data
// round-trip. Tracked by ASYNCcnt.
__device__ __forceinline__ void async_copy_b128(unsigned int lds_addr,
                                                const float* gaddr) {
  asm volatile("global_load_async_to_lds_b128 %0, %1, off"
               :
               : "v"(lds_addr), "v"(gaddr)
               : "memory");
}

// Async loads complete in order, so "<= 8 outstanding" proves the
// previous 8-op batch has fully landed while the newest batch is in flight.
__device__ __forceinline__ void wait_async_le8() {
#if __has_builtin(__builtin_amdgcn_s_wait_asynccnt)
  __builtin_amdgcn_s_wait_asynccnt(8);
#else
  asm volatile("s_wait_asynccnt 0x8" ::: "memory");
#endif
}

__device__ __forceinline__ void wait_async_zero() {
#if __has_builtin(__builtin_amdgcn_s_wait_asynccnt)
  __builtin_amdgcn_s_wait_asynccnt(0);
#else
  asm volatile("s_wait_asynccnt 0x0" ::: "memory");
#endif
}

// ---------------------------------------------------------------------------
// Phase 1: W_comb[d, b] = W_aff[b, d] + w_block0[f0[d], b] + w_block1[f1[d], b]
// LDS-tiled transpose: coalesced reads of W_aff along d, coalesced writes
// along b. f0[d]/f1[d] are wave-uniform (scalar loads), block rows coalesced.
// ---------------------------------------------------------------------------
__global__ __launch_bounds__(256) void combine_kernel(
    const float* __restrict__ W_aff, const float* __restrict__ w0,
    const float* __restrict__ w1, const int* __restrict__ f0,
    const int* __restrict__ f1, float* __restrict__ W_comb) {
  __shared__ float tile[32][33];
  const int d0 = blockIdx.x * 32;
  const int b0 = blockIdx.y * 32;
  const int tx = threadIdx.x;  // 0..31
  const int ty = threadIdx.y;  // 0..7
#pragma unroll
  for (int i = 0; i < 32; i += 8)
    tile[ty + i][tx] = W_aff[(size_t)(b0 + ty + i) * DFEAT + (d0 + tx)];
  __syncthreads();
#pragma unroll
  for (int i = 0; i < 32; i += 8) {
    const int d = d0 + ty + i;
    const int b = b0 + tx;
    float v = tile[tx][ty + i];
    v += w0[(size_t)f0[d] * BASEDIM + b];
    v += w1[(size_t)f1[d] * BASEDIM + b];
    W_comb[(size_t)d * BASEDIM + b] = v;
  }
}

// ---------------------------------------------------------------------------
// Phase 2: out[n, :] = b_aff + sum_k v[n,k] * W_comb[idx[n,k], :]
// 16 positions per workgroup; K-dim = 512 (n,k) slots, A block-diagonal.
// One pipeline stage = one position (32 gathered rows = 32 KB per column
// half), triple-buffered in LDS with depth-2 global_load_async_to_lds_b128
// prefetch: each DMA batch has two full compute stages of latency slack.
// Compute per stage per wave: 8 x V_WMMA_F32_16X16X4_F32 x 2 column tiles.
// ---------------------------------------------------------------------------
__global__ __launch_bounds__(256) void ft_wmma_kernel(
    const float* __restrict__ x_val, const float* __restrict__ b_aff,
    const int* __restrict__ x_idx, const float* __restrict__ W_comb,
    float* __restrict__ out) {
  __shared__ float vtab[16][KACT];
  __shared__ int   itab[16][KACT];
  __shared__ __align__(16) float Bn[NBUF][KACT][BSTRIDE];  // [buf][k][column]

  const int g    = blockIdx.x;   // n-group: rows g*16 .. g*16+15
  const int cy   = blockIdx.y;   // column half
  const int t    = threadIdx.x;  // 0..255
  const int lane = t & 31;
  const int wave = t >> 5;       // 0..7
  const int hi   = lane >> 4;    // half-wave: 0 or 1
  const int m    = lane & 15;    // row (A) / column (B/C) within tile

  // Per-thread staging slot: base row of the stage, 16-byte column quad.
  const int sr    = t >> 6;        // 0..3
  const int scolq = (t & 63) * 4;  // 0,4,...,252

  const int col0 = (wave * 2) * 16 + m;        // this wave's B/C columns
  const int col1 = (wave * 2 + 1) * 16 + m;
  const float* Wbase = W_comb + (size_t)cy * HALFCOLS + scolq;

  // Stage per-position values and indices.
  for (int i = t; i < 16 * KACT; i += 256) {
    const int n = i >> 5, k = i & 31;
    vtab[n][k] = x_val[(size_t)(g * 16 + n) * KACT + k];
    itab[n][k] = x_idx[(size_t)(g * 16 + n) * KACT + k];
  }
  __syncthreads();

  // Issue 8 async b128 copies covering one position's 32 rows (this thread's
  // rows: sr, sr+4, ..., sr+28).
  auto stage = [&](int n, int buf) {
#pragma unroll
    for (int j = 0; j < 8; ++j) {
      const int rk  = j * 4 + sr;
      const int row = itab[n][rk];
      async_copy_b128((unsigned int)(size_t)&Bn[buf][rk][scolq],
                      Wbase + (size_t)row * BASEDIM);
    }
  };

  v8f acc0 = {};
  v8f acc1 = {};

  auto compute = [&](int n, int buf) {
    const bool mine = (m == n);
#pragma unroll
    for (int j = 0; j < 8; ++j) {
      const int kb = 4 * j;
      // A 16x4 f32 layout: lane holds row M = lane&15; VGPR0 = K(2*hi),
      // VGPR1 = K(2*hi+1). Block-diagonal: nonzero only on row n.
      const float av0 = vtab[n][kb + 2 * hi];
      const float av1 = vtab[n][kb + 2 * hi + 1];
      v2f A;
      A.x = mine ? av0 : 0.0f;
      A.y = mine ? av1 : 0.0f;
      // B 4x16 f32 layout: lane = column, VGPR0 = row kb+2*hi, VGPR1 = +1.
      {
        v2f B;
        B.x = Bn[buf][kb + 2 * hi][col0];
        B.y = Bn[buf][kb + 2 * hi + 1][col0];
        acc0 = __builtin_amdgcn_wmma_f32_16x16x4_f32(
            false, A, false, B, (short)0, acc0, false, false);
      }
      {
        v2f B;
        B.x = Bn[buf][kb + 2 * hi][col1];
        B.y = Bn[buf][kb + 2 * hi + 1][col1];
        acc1 = __builtin_amdgcn_wmma_f32_16x16x4_f32(
            false, A, false, B, (short)0, acc1, false, false);
      }
    }
  };

  // Prime the pipeline two stages deep.
  stage(0, 0);
  stage(1, 1);
  wait_async_le8();  // stage 0 landed; stage 1 still in flight
  __syncthreads();

  int cbuf = 0;  // buffer holding stage s
  int nbuf = 2;  // buffer for stage s+2
  for (int s = 0; s < 16; ++s) {
    if (s + 2 < 16) stage(s + 2, nbuf);
    compute(s, cbuf);
    if (s + 2 < 16)
      wait_async_le8();   // stage s+1 landed; stage s+2 in flight
    else
      wait_async_zero();  // drain the tail
    __syncthreads();
    cbuf = (cbuf == NBUF - 1) ? 0 : cbuf + 1;
    nbuf = (nbuf == NBUF - 1) ? 0 : nbuf + 1;
  }

  // C/D 16x16 f32 layout: VGPR r holds M = 8*hi + r, column = lane&15.
#pragma unroll
  for (int r = 0; r < 8; ++r) {
    const int row_n = g * 16 + hi * 8 + r;
    const int oc0   = cy * HALFCOLS + col0;
    const int oc1   = cy * HALFCOLS + col1;
    out[(size_t)row_n * BASEDIM + oc0] = acc0[r] + b_aff[oc0];
    out[(size_t)row_n * BASEDIM + oc1] = acc1[r] + b_aff[oc1];
  }
}

extern "C" void kernel_launch(void* const* d_in, const int* in_sizes, int n_in,
                              void* d_out, int out_size, void* d_ws,
                              size_t ws_size, hipStream_t stream) {
  const float* x_val = (const float*)d_in[0];  // [N, K]
  const float* W_aff = (const float*)d_in[1];  // [BASE, D]
  const float* b_aff = (const float*)d_in[2];  // [BASE]
  const float* w0    = (const float*)d_in[3];  // [F0, BASE]
  const float* w1    = (const float*)d_in[4];  // [F1, BASE]
  const int*   x_idx = (const int*)d_in[5];    // [N, K]
  const int*   f0    = (const int*)d_in[6];    // [D]
  const int*   f1    = (const int*)d_in[7];    // [D]
  float*       outp  = (float*)d_out;          // [N, BASE]
  float*       W_comb = (float*)d_ws;          // [D, BASE] = 96 MB scratch

  // Phase 1: build fused gather table (transpose + factored-block fold-in).
  dim3 gridC(DFEAT / 32, BASEDIM / 32);
  dim3 blockC(32, 8);
  combine_kernel<<<gridC, blockC, 0, stream>>>(W_aff, w0, w1, f0, f1, W_comb);

  // Phase 2: gather-accumulate via WMMA, triple-buffered async-to-LDS.
  dim3 gridM(N_POS / 16, CSPLIT);
  ft_wmma_kernel<<<gridM, 256, 0, stream>>>(x_val, b_aff, x_idx, W_comb, outp);
}